// MS_Part_SSIM_25864293056956
// MI455X (gfx1250) — compile-verified
//
#include <hip/hip_runtime.h>
#include <hip/hip_bf16.h>
#include <cstdint>

typedef __attribute__((ext_vector_type(2))) float v2f;
typedef __attribute__((ext_vector_type(4))) float v4f;
typedef __attribute__((ext_vector_type(8))) float v8f;

// Types matching the async-copy builtin's parameters (from hipcc diagnostic:
// param0 is "int __attribute__((vector_size(16))) __device__ *", i.e. AS1 int4*)
typedef int v4i_n __attribute__((vector_size(16)));
typedef __attribute__((address_space(1))) v4i_n as1_v4i;
typedef __attribute__((address_space(3))) v4i_n as3_v4i;

#define Bn 4
#define Cn 3
#define Hn 256
#define Wn 176
#define HW (Hn * Wn)             // 45056
#define Sn 5
#define Mn 18
#define SM (Sn * Mn)             // 90
#define ROWTILES 6               // 96 padded rows
#define KCHUNKS 88               // 45056 / 88 = 512 exactly
#define KPER (HW / KCHUNKS)      // 512
#define NWAVES 8                 // 256 threads = 8 wave32
#define KPERWAVE (KPER / NWAVES) // 64
#define LDS_B_FLOATS (KPER * 16)                 // 8192 floats (32 KB)
#define LDS_RED_FLOATS (NWAVES * ROWTILES * 256) // 12288 floats (48 KB)
#define LDS_FLOATS (LDS_RED_FLOATS)              // max of the two uses

// ---------------------------------------------------------------------------
// CDNA5 WMMA f32 16x16x4 (codegen-confirmed in round 1).
// ---------------------------------------------------------------------------
static __device__ __forceinline__ v8f wmma_16x16x4_f32(v2f a, v2f b, v8f c) {
#if __has_builtin(__builtin_amdgcn_wmma_f32_16x16x4_f32)
  return __builtin_amdgcn_wmma_f32_16x16x4_f32(false, a, false, b, (short)0, c,
                                               false, false);
#else
  int lane = (int)(threadIdx.x & 31u);
  int n16 = lane & 15;
  int hL = lane >> 4;
#pragma unroll
  for (int r = 0; r < 8; ++r) {
    int m = r + 8 * hL;
    float d = c[r];
#pragma unroll
    for (int k = 0; k < 4; ++k) {
      float av = __shfl((k & 1) ? a.y : a.x, ((k >= 2) ? 16 : 0) + m, 32);
      float bv = __shfl((k & 1) ? b.y : b.x, ((k >= 2) ? 16 : 0) + n16, 32);
      d += av * bv;
    }
    c[r] = d;
  }
  return c;
#endif
}

// ---------------------------------------------------------------------------
// Async global->LDS 128b copy (gfx1250 GLOBAL_LOAD_ASYNC_TO_LDS_B128,
// ASYNCcnt-tracked). Pointers built via uintptr_t casts into properly
// typedef'd address-space pointers (generic-LDS addr low 32 bits == LDS off).
// ---------------------------------------------------------------------------
#if defined(__has_builtin)
#if __has_builtin(__builtin_amdgcn_global_load_async_to_lds_b128)
#define ASYNC_G2L 1
#endif
#endif
#ifndef ASYNC_G2L
#define ASYNC_G2L 0
#endif

static __device__ __forceinline__ void async_g2l_b128(const void* g, void* l) {
#if ASYNC_G2L
  __builtin_amdgcn_global_load_async_to_lds_b128(
      (as1_v4i*)(uintptr_t)g, (as3_v4i*)(uintptr_t)l, 0, 0);
#else
  *(v4f*)l = *(const v4f*)g;
#endif
}

static __device__ __forceinline__ void async_wait0() {
#if ASYNC_G2L
#if __has_builtin(__builtin_amdgcn_s_wait_asynccnt)
  __builtin_amdgcn_s_wait_asynccnt(0);
#else
  asm volatile("s_wait_asynccnt 0" ::: "memory");
#endif
#endif
}

// ---------------------------------------------------------------------------
// Kernel 1: build B-panel P[b][k][16]:
//   cols c*5+{0..4} = Xh, Yh, Xh^2, Yh^2, Xh*Yh (Xh=(X+1)/2), col 15 = 1.
// The ones-column makes msum fall out of the same single GEMM pass.
// ---------------------------------------------------------------------------
__global__ void build_panel_kernel(const float* __restrict__ X,
                                   const float* __restrict__ Y,
                                   float* __restrict__ P) {
  int idx = blockIdx.x * blockDim.x + threadIdx.x;
  if (idx >= Bn * HW) return;
  int b = idx / HW;
  int k = idx - b * HW;

  float col[16];
#pragma unroll
  for (int c = 0; c < Cn; ++c) {
    float x = X[(size_t)(b * Cn + c) * HW + k];
    float y = Y[(size_t)(b * Cn + c) * HW + k];
    x = (x + 1.0f) * 0.5f;
    y = (y + 1.0f) * 0.5f;
    col[c * 5 + 0] = x;
    col[c * 5 + 1] = y;
    col[c * 5 + 2] = x * x;
    col[c * 5 + 3] = y * y;
    col[c * 5 + 4] = x * y;
  }
  col[15] = 1.0f;

  v4f* out = reinterpret_cast<v4f*>(P + (size_t)idx * 16);
#pragma unroll
  for (int q = 0; q < 4; ++q) {
    v4f t = {col[q * 4 + 0], col[q * 4 + 1], col[q * 4 + 2], col[q * 4 + 3]};
    out[q] = t;
  }
}

// ---------------------------------------------------------------------------
// Kernel 2: one block per (b, k-chunk). Async-stage the 32 KB B chunk into
// LDS once; each wave accumulates ALL 6 row tiles against it (B fragment
// loaded once per k-step, 6 WMMAs issued against it). Cross-wave reduction
// reuses the LDS region after a barrier. Deterministic partials to d_ws.
// ---------------------------------------------------------------------------
__global__ void __launch_bounds__(256) moment_gemm_kernel(
    const float* __restrict__ mask, const float* __restrict__ P,
    float* __restrict__ partial) {
  __shared__ float smem[LDS_FLOATS];

  int bId = blockIdx.x;
  int chunk = bId % KCHUNKS;
  int b = bId / KCHUNKS;

  int tid = (int)threadIdx.x;
  int wave = tid >> 5;
  int lane = tid & 31;
  int lane16 = lane & 15;
  int half = lane >> 4;

  // ---- stage B chunk (KPER x 16 f32 = 32 KB contiguous) into LDS ----
  {
    const char* src =
        (const char*)(P + ((size_t)b * HW + (size_t)chunk * KPER) * 16);
    char* dst = (char*)smem;
#pragma unroll
    for (int i = 0; i < (KPER * 16 * 4) / (256 * 16); ++i) {  // 8 iters
      int off = (i * 256 + tid) * 16;
      async_g2l_b128(src + off, dst + off);
    }
  }

  // ---- per-tile A row pointers (M = lane%16 per the ISA A layout) ----
  const float* aP[ROWTILES];
  float sc[ROWTILES];
#pragma unroll
  for (int t = 0; t < ROWTILES; ++t) {
    int row = t * 16 + lane16;
    int rowc = (row < SM) ? row : 0;
    aP[t] = mask + (size_t)(b * SM + rowc) * HW;
    sc[t] = (row < SM) ? 1.0f : 0.0f;
  }

  int kBase = chunk * KPER + wave * KPERWAVE;  // global k for A loads
  int kLds = wave * KPERWAVE;                  // k within LDS B stage

  // gfx1250 prefetch (speculative): warm the A stream for this wave
#pragma unroll
  for (int t = 0; t < ROWTILES; ++t)
    __builtin_prefetch(aP[t] + kBase, 0, 0);

  v8f acc[ROWTILES];
#pragma unroll
  for (int t = 0; t < ROWTILES; ++t) {
    v8f z = {0.f, 0.f, 0.f, 0.f, 0.f, 0.f, 0.f, 0.f};
    acc[t] = z;
  }

  async_wait0();
  __syncthreads();

  for (int kk = 0; kk < KPERWAVE; kk += 4) {
    int klA = kBase + kk + 2 * half;  // per-lane K base (lanes 0-15: K0,1;
    int klB = kLds + kk + 2 * half;   //                  lanes 16-31: K2,3)
    v2f bb;
    bb.x = smem[klB * 16 + lane16];
    bb.y = smem[(klB + 1) * 16 + lane16];
#pragma unroll
    for (int t = 0; t < ROWTILES; ++t) {
      v2f a = *reinterpret_cast<const v2f*>(aP[t] + klA);
      a.x *= sc[t];
      a.y *= sc[t];
      acc[t] = wmma_16x16x4_f32(a, bb, acc[t]);
    }
  }

  __syncthreads();  // all waves done reading the B stage; reuse LDS
#pragma unroll
  for (int t = 0; t < ROWTILES; ++t) {
#pragma unroll
    for (int r = 0; r < 8; ++r)
      smem[(wave * ROWTILES + t) * 256 + (r + 8 * half) * 16 + lane16] =
          acc[t][r];
  }
  __syncthreads();

#pragma unroll
  for (int t = 0; t < ROWTILES; ++t) {
    float s = 0.f;
#pragma unroll
    for (int w = 0; w < NWAVES; ++w) s += smem[(w * ROWTILES + t) * 256 + tid];
    partial[(((size_t)chunk * Bn + b) * ROWTILES + t) * 256 + tid] = s;
  }
}

// ---------------------------------------------------------------------------
// Kernel 3: deterministic reduction over K-chunks -> mom[b][96][16]
// ---------------------------------------------------------------------------
__global__ void reduce_partials_kernel(const float* __restrict__ partial,
                                       float* __restrict__ mom) {
  int bt = blockIdx.x;  // b*ROWTILES + tile, 0..23
  int b = bt / ROWTILES;
  int t = bt % ROWTILES;
  int tid = (int)threadIdx.x;  // 0..255 -> (r, n) of the tile
  float s = 0.f;
  for (int c = 0; c < KCHUNKS; ++c)
    s += partial[(((size_t)c * Bn + b) * ROWTILES + t) * 256 + tid];
  mom[(size_t)bt * 256 + tid] = s;  // == mom[(b*96 + row)*16 + col]
}

// ---------------------------------------------------------------------------
// Kernel 4: SSIM epilogue (normalize, SSIM/CS, relu, means, scale product,
// batch mean -> scalar).
// ---------------------------------------------------------------------------
__global__ void ssim_final_kernel(const float* __restrict__ mom,
                                  float* __restrict__ out) {
  __shared__ float csb[Bn * Sn];
  __shared__ float ssb[Bn * Sn];
  int t = (int)threadIdx.x;

  if (t < Bn * Sn) {
    int b = t / Sn;
    int s = t % Sn;
    const float C1 = 1e-4f;  // (0.01*1.0)^2
    const float C2 = 9e-4f;  // (0.03*1.0)^2
    const float EPSf = 1e-6f;
    float cs_acc = 0.f, ss_acc = 0.f;
    for (int m = 0; m < Mn; ++m) {
      int row = s * Mn + m;
      const float* v = mom + ((size_t)b * (ROWTILES * 16) + row) * 16;
      float inv = 1.0f / (v[15] + EPSf);  // msum + eps
      for (int c = 0; c < Cn; ++c) {
        float mu1 = v[c * 5 + 0] * inv;
        float mu2 = v[c * 5 + 1] * inv;
        float mu11 = v[c * 5 + 2] * inv;
        float mu22 = v[c * 5 + 3] * inv;
        float mu12 = v[c * 5 + 4] * inv;
        float mu1sq = mu1 * mu1;
        float mu2sq = mu2 * mu2;
        float mu1mu2 = mu1 * mu2;
        float s1 = mu11 - mu1sq;
        float s2 = mu22 - mu2sq;
        float s12 = mu12 - mu1mu2;
        float cs = (2.0f * s12 + C2) / (s1 + s2 + C2);
        float ssim = ((2.0f * mu1mu2 + C1) / (mu1sq + mu2sq + C1)) * cs;
        cs_acc += fmaxf(cs, 0.0f);
        ss_acc += fmaxf(ssim, 0.0f);
      }
    }
    csb[t] = cs_acc / (float)(Mn * Cn);
    ssb[t] = ss_acc / (float)(Mn * Cn);
  }
  __syncthreads();

  if (t == 0) {
    float acc = 0.f;
    for (int b = 0; b < Bn; ++b) {
      float p = ssb[b * Sn + (Sn - 1)];
      for (int s = 0; s < Sn - 1; ++s) p *= csb[b * Sn + s];
      acc += p;
    }
    out[0] = acc / (float)Bn;
  }
}

// ---------------------------------------------------------------------------
extern "C" void kernel_launch(void* const* d_in, const int* in_sizes, int n_in,
                              void* d_out, int out_size, void* d_ws,
                              size_t ws_size, hipStream_t stream) {
  const float* X = (const float*)d_in[0];     // [4,3,256,176]
  const float* Y = (const float*)d_in[1];     // [4,3,256,176]
  const float* mask = (const float*)d_in[2];  // [4,5,18,256,176]

  float* P = (float*)d_ws;                    // 4*45056*16 floats (11.5 MB)
  float* partial = P + (size_t)Bn * HW * 16;  // 88*4*6*256 floats (2.1 MB)
  float* mom = partial + (size_t)KCHUNKS * Bn * ROWTILES * 256;  // 6144 floats

  build_panel_kernel<<<(Bn * HW + 255) / 256, 256, 0, stream>>>(X, Y, P);
  moment_gemm_kernel<<<KCHUNKS * Bn, 256, 0, stream>>>(mask, P, partial);
  reduce_partials_kernel<<<Bn * ROWTILES, 256, 0, stream>>>(partial, mom);
  ssim_final_kernel<<<1, 64, 0, stream>>>(mom, (float*)d_out);
}